// fusion_model_mae_2_21689584844957
// MI455X (gfx1250) — compile-verified
//
#include <hip/hip_runtime.h>

// ---------------------------------------------------------------------------
// GCN-JK fusion for MI455X (gfx1250, wave32).
// GEMMs use V_WMMA_F32_16X16X4_F32 (exact fp32). Edge aggregation uses
// float4 gathers + global_atomic_add_f32 scatters (memory-bound part).
// ---------------------------------------------------------------------------

typedef __attribute__((ext_vector_type(2))) float v2f;
typedef __attribute__((ext_vector_type(8))) float v8f;

#define FDIM   128          // hidden feature dim (fixed by the model)
#define FSH    7            // log2(FDIM)
#define CH     32           // float4 chunks per row = FDIM/4
#define CSH    5            // log2(CH)

// ---------------- utility ----------------
__global__ void zero_f32(float* __restrict__ p, long n) {
    long i = (long)blockIdx.x * blockDim.x + threadIdx.x;
    if (i < n) p[i] = 0.0f;
}

// degin[col]++ ; degout[row]++  (float accumulators)
__global__ void edge_degrees(const int* __restrict__ row, const int* __restrict__ col,
                             float* __restrict__ degin, float* __restrict__ degout, int E) {
    int e = blockIdx.x * blockDim.x + threadIdx.x;
    if (e < E) {
        atomicAdd(&degin[col[e]], 1.0f);
        atomicAdd(&degout[row[e]], 1.0f);
    }
}

// dinv[i] = rsqrt(in_degree + 1)   (+1 = self loop; always > 0)
__global__ void finalize_dinv(float* __restrict__ dinv, int n) {
    int i = blockIdx.x * blockDim.x + threadIdx.x;
    if (i < n) dinv[i] = rsqrtf(dinv[i] + 1.0f);
}

// ---------------- fp32 WMMA GEMM: C[n x fout] = A[n x fin] @ W[fin x fout] (+bias) ----
// One wave computes one 16x16 tile; blockIdx.x = row tile, wave id = col tile.
// Requires n % 16 == 0 (N = 50000 = 3125 * 16) so EXEC is all-1s at every WMMA.
__global__ void __launch_bounds__(256)
gemm_wmma_f32(const float* __restrict__ A, const float* __restrict__ W,
              const float* __restrict__ bias, float* __restrict__ C,
              int fin, int fout) {
    const int lane = threadIdx.x & 31;
    const int wv   = threadIdx.x >> 5;
    const int tm   = blockIdx.x << 4;
    const int tn   = wv << 4;
    const int l15  = lane & 15;
    const bool hi  = lane >= 16;           // lanes 16-31 carry K+2,K+3 halves
    const int m    = tm + l15;
    const int n    = tn + l15;

    const float* __restrict__ arow = A + (long)m * fin + (hi ? 2 : 0);
    const float* __restrict__ bcol = W + n + (long)(hi ? 2 : 0) * fout;

    v8f acc = 0.0f;
    for (int k = 0; k < fin; k += 4) {
        v2f a, b;
        a.x = arow[k];
        a.y = arow[k + 1];
        b.x = bcol[(long)k * fout];
        b.y = bcol[(long)(k + 1) * fout];
        acc = __builtin_amdgcn_wmma_f32_16x16x4_f32(
                  /*neg_a=*/false, a, /*neg_b=*/false, b,
                  /*c_mod=*/(short)0, acc, /*reuse_a=*/false, /*reuse_b=*/false);
    }

    const float bv = bias ? bias[n] : 0.0f;
    const int rbase = tm + (hi ? 8 : 0);
#pragma unroll
    for (int j = 0; j < 8; ++j) {
        C[(long)(rbase + j) * fout + n] = acc[j] + bv;
    }
}

// ---------------- GCN aggregation ----------------
// out[i][:] = h[i][:] * dinv[i]^2   (self-loop term; also initializes out)
__global__ void selfloop_init(float* __restrict__ out, const float* __restrict__ h,
                              const float* __restrict__ dinv, long nchunks) {
    long i = (long)blockIdx.x * blockDim.x + threadIdx.x;   // over N*CH
    if (i >= nchunks) return;
    const long node = i >> CSH;
    float d = dinv[node]; d = d * d;
    float4 v = ((const float4*)h)[i];
    float4 o = make_float4(v.x * d, v.y * d, v.z * d, v.w * d);
    ((float4*)out)[i] = o;
}

// out[c][:] += h[r][:] * dinv[r]*dinv[c]   for every edge
__global__ void gcn_edge(float* __restrict__ out, const float* __restrict__ h,
                         const int* __restrict__ row, const int* __restrict__ col,
                         const float* __restrict__ dinv, long nthreads) {
    long i = (long)blockIdx.x * blockDim.x + threadIdx.x;   // over E*CH
    if (i >= nthreads) return;
    const long e  = i >> CSH;
    const int  ck = (int)(i & (CH - 1));
    const int  r  = row[e];
    const int  c  = col[e];
    const float norm = dinv[r] * dinv[c];
    float4 v = ((const float4*)(h + ((long)r << FSH)))[ck];
    float* dst = out + ((long)c << FSH) + (ck << 2);
    atomicAdd(dst + 0, v.x * norm);
    atomicAdd(dst + 1, v.y * norm);
    atomicAdd(dst + 2, v.z * norm);
    atomicAdd(dst + 3, v.w * norm);
}

// out[i] = max(out[i] + b[i % FDIM], 0)
__global__ void bias_relu(float* __restrict__ out, const float* __restrict__ b, long n) {
    long i = (long)blockIdx.x * blockDim.x + threadIdx.x;
    if (i < n) out[i] = fmaxf(out[i] + b[i & (FDIM - 1)], 0.0f);
}

// ---------------- neighbor mean ----------------
// s[r][:] += h[c][:]
__global__ void nm_edge(float* __restrict__ s, const float* __restrict__ h,
                        const int* __restrict__ row, const int* __restrict__ col,
                        long nthreads) {
    long i = (long)blockIdx.x * blockDim.x + threadIdx.x;   // over E*CH
    if (i >= nthreads) return;
    const long e  = i >> CSH;
    const int  ck = (int)(i & (CH - 1));
    const int  r  = row[e];
    const int  c  = col[e];
    float4 v = ((const float4*)(h + ((long)c << FSH)))[ck];
    float* dst = s + ((long)r << FSH) + (ck << 2);
    atomicAdd(dst + 0, v.x);
    atomicAdd(dst + 1, v.y);
    atomicAdd(dst + 2, v.z);
    atomicAdd(dst + 3, v.w);
}

// dst[i] = deg>0 ? s[i]/deg : h[i]
__global__ void nm_finalize(float* __restrict__ dst, const float* __restrict__ s,
                            const float* __restrict__ h, const float* __restrict__ degout,
                            long nchunks) {
    long i = (long)blockIdx.x * blockDim.x + threadIdx.x;   // over N*CH
    if (i >= nchunks) return;
    const long node = i >> CSH;
    const float deg = degout[node];
    float4 o;
    if (deg > 0.0f) {
        const float inv = 1.0f / deg;
        float4 v = ((const float4*)s)[i];
        o = make_float4(v.x * inv, v.y * inv, v.z * inv, v.w * inv);
    } else {
        o = ((const float4*)h)[i];
    }
    ((float4*)dst)[i] = o;
}

// ---------------- JK combine: dst = w0*h0 + w1*h1 + w2*h2 (softmax weights) --------
__global__ void jk_combine(float* __restrict__ dst, const float* __restrict__ h0,
                           const float* __restrict__ h1, const float* __restrict__ h2,
                           const float* __restrict__ jkw, long nchunks) {
    long i = (long)blockIdx.x * blockDim.x + threadIdx.x;   // over N*CH
    if (i >= nchunks) return;
    const float a = jkw[0], b = jkw[1], c = jkw[2];
    const float mx = fmaxf(a, fmaxf(b, c));
    const float e0 = __expf(a - mx), e1 = __expf(b - mx), e2 = __expf(c - mx);
    const float inv = 1.0f / (e0 + e1 + e2);
    const float w0 = e0 * inv, w1 = e1 * inv, w2 = e2 * inv;
    float4 v0 = ((const float4*)h0)[i];
    float4 v1 = ((const float4*)h1)[i];
    float4 v2 = ((const float4*)h2)[i];
    float4 o = make_float4(w0 * v0.x + w1 * v1.x + w2 * v2.x,
                           w0 * v0.y + w1 * v1.y + w2 * v2.y,
                           w0 * v0.z + w1 * v1.z + w2 * v2.z,
                           w0 * v0.w + w1 * v1.w + w2 * v2.w);
    ((float4*)dst)[i] = o;
}

// ---------------------------------------------------------------------------
static inline int cdiv(long a, long b) { return (int)((a + b - 1) / b); }

extern "C" void kernel_launch(void* const* d_in, const int* in_sizes, int n_in,
                              void* d_out, int out_size, void* d_ws, size_t ws_size,
                              hipStream_t stream) {
    const float* x   = (const float*)d_in[0];
    const int*   ei  = (const int*)d_in[1];
    const float* W1  = (const float*)d_in[2];
    const float* b1  = (const float*)d_in[3];
    const float* W2  = (const float*)d_in[4];
    const float* b2  = (const float*)d_in[5];
    const float* W3  = (const float*)d_in[6];
    const float* b3  = (const float*)d_in[7];
    const float* jkw = (const float*)d_in[8];
    const float* fcW = (const float*)d_in[9];
    const float* fcb = (const float*)d_in[10];
    float* out = (float*)d_out;

    const int  F    = FDIM;
    const int  OUTF = 64;
    const int  N    = in_sizes[0] / F;      // 50000 (multiple of 16)
    const int  E    = in_sizes[1] / 2;      // 800000
    const int* row  = ei;                   // edge_index[0]
    const int* col  = ei + E;               // edge_index[1]

    float* ws = (float*)d_ws;
    const long NF = (long)N * F;
    float* bufA   = ws;             // lin scratch / nm-sum / JK result
    float* bufB   = ws + 1 * NF;    // h0, later h1
    float* bufC   = ws + 2 * NF;    // h0 after neighbor-mean (kept for JK)
    float* bufD   = ws + 3 * NF;    // h2 (kept for JK)
    float* dinv   = ws + 4 * NF;    // [N]  rsqrt(in-degree + 1)
    float* degout = dinv + N;       // [N]  out-degree (for neighbor mean)

    const long nodeChunks = (long)N * CH;   // N * 32 float4 chunks
    const long edgeThr    = (long)E * CH;   // E * 32
    const int  rowTiles   = N / 16;         // 3125

    // ---- degrees & normalization ----
    zero_f32<<<cdiv(2L * N, 256), 256, 0, stream>>>(dinv, 2L * N);
    edge_degrees<<<cdiv(E, 256), 256, 0, stream>>>(row, col, dinv, degout, E);
    finalize_dinv<<<cdiv(N, 256), 256, 0, stream>>>(dinv, N);

    // ---- layer 0: GCN(x, W1) -> ReLU ----
    gemm_wmma_f32<<<rowTiles, (F / 16) * 32, 0, stream>>>(x, W1, nullptr, bufA, F, F);
    selfloop_init<<<cdiv(nodeChunks, 256), 256, 0, stream>>>(bufB, bufA, dinv, nodeChunks);
    gcn_edge<<<cdiv(edgeThr, 256), 256, 0, stream>>>(bufB, bufA, row, col, dinv, edgeThr);
    bias_relu<<<cdiv(NF, 256), 256, 0, stream>>>(bufB, b1, NF);

    // ---- neighbor mean: h0 = mean_{c in out(r)} h0[c], fallback h0 if deg==0 ----
    zero_f32<<<cdiv(NF, 256), 256, 0, stream>>>(bufA, NF);
    nm_edge<<<cdiv(edgeThr, 256), 256, 0, stream>>>(bufA, bufB, row, col, edgeThr);
    nm_finalize<<<cdiv(nodeChunks, 256), 256, 0, stream>>>(bufC, bufA, bufB, degout, nodeChunks);

    // ---- layer 1: GCN(h0m, W2) -> ReLU  (h1 in bufB) ----
    gemm_wmma_f32<<<rowTiles, (F / 16) * 32, 0, stream>>>(bufC, W2, nullptr, bufA, F, F);
    selfloop_init<<<cdiv(nodeChunks, 256), 256, 0, stream>>>(bufB, bufA, dinv, nodeChunks);
    gcn_edge<<<cdiv(edgeThr, 256), 256, 0, stream>>>(bufB, bufA, row, col, dinv, edgeThr);
    bias_relu<<<cdiv(NF, 256), 256, 0, stream>>>(bufB, b2, NF);

    // ---- layer 2: GCN(h1, W3) -> ReLU  (h2 in bufD) ----
    gemm_wmma_f32<<<rowTiles, (F / 16) * 32, 0, stream>>>(bufB, W3, nullptr, bufA, F, F);
    selfloop_init<<<cdiv(nodeChunks, 256), 256, 0, stream>>>(bufD, bufA, dinv, nodeChunks);
    gcn_edge<<<cdiv(edgeThr, 256), 256, 0, stream>>>(bufD, bufA, row, col, dinv, edgeThr);
    bias_relu<<<cdiv(NF, 256), 256, 0, stream>>>(bufD, b3, NF);

    // ---- JK: softmax(jk_w)-weighted sum of h0m, h1, h2 ----
    jk_combine<<<cdiv(nodeChunks, 256), 256, 0, stream>>>(bufA, bufC, bufB, bufD, jkw, nodeChunks);

    // ---- fc: out = hjk @ fc_W + fc_b ----
    gemm_wmma_f32<<<rowTiles, (OUTF / 16) * 32, 0, stream>>>(bufA, fcW, fcb, out, F, OUTF);
}